// HCE_Linear_28200755265773
// MI455X (gfx1250) — compile-verified
//
#include <hip/hip_runtime.h>

// MoE hard-routing forward: y[b] = x[b] @ W[argmax_e sigmoid(x@Wp.T+bp)]
// B=4096, I=256, O=256, E=8.  fp32 end-to-end using V_WMMA_F32_16X16X4_F32.

typedef __attribute__((ext_vector_type(2))) float v2f;
typedef __attribute__((ext_vector_type(8))) float v8f;

#define NB 4096
#define NI 256
#define NO 256
#define NE 8
#define TIE_FLAG (1 << 30)
#define ROWS_PB 32                 // gathered rows per GEMM block (2 M-tiles)

// ---------------------------------------------------------------- kernel 0
__global__ void zero_counts(int* __restrict__ counts) {
    if (threadIdx.x < NE) counts[threadIdx.x] = 0;
}

// ---------------------------------------------------------------- kernel 1
// One block = 64 rows, 256 threads. Thread layout: r = tid%64 row, p = tid/64
// covers 16 columns of each 64-column chunk. Partial dots reduced via LDS.
__global__ __launch_bounds__(256) void router_kernel(
    const float* __restrict__ x, const float* __restrict__ Wp,
    const float* __restrict__ bp, int* __restrict__ counts,
    int* __restrict__ list, float* __restrict__ out) {
    __shared__ float wps[NE * NI];       // 8KB
    __shared__ float xs[64 * 65];        // 16.25KB, pitch 65 avoids conflicts
    __shared__ float part[4][64][NE];    // 8KB

    const int tid = threadIdx.x;
    const int rb  = blockIdx.x * 64;

    // stage Wp (8x256) into LDS, coalesced
    #pragma unroll
    for (int j = 0; j < 8; j++) wps[tid + j * 256] = Wp[tid + j * 256];

    const int r = tid & 63;
    const int p = tid >> 6;
    float acc[NE];
    #pragma unroll
    for (int e = 0; e < NE; e++) acc[e] = 0.f;

    for (int chunk = 0; chunk < 4; chunk++) {
        const int c0 = chunk * 64;
        __syncthreads();  // protect xs reuse (and wps on first pass)
        // cooperative coalesced load: 64 rows x 64 cols = 1024 float4
        for (int q = tid; q < 1024; q += 256) {
            const int rr = q >> 4;        // 16 float4 per row-chunk
            const int cw = q & 15;
            const float4 v =
                *(const float4*)(x + (size_t)(rb + rr) * NI + c0 + cw * 4);
            float* d = &xs[rr * 65 + cw * 4];
            d[0] = v.x; d[1] = v.y; d[2] = v.z; d[3] = v.w;
        }
        __syncthreads();
        #pragma unroll
        for (int i = 0; i < 16; i++) {
            const float xv = xs[r * 65 + p * 16 + i];
            const int   cc = c0 + p * 16 + i;
            #pragma unroll
            for (int e = 0; e < NE; e++) acc[e] += xv * wps[e * 256 + cc];
        }
    }
    #pragma unroll
    for (int e = 0; e < NE; e++) part[p][r][e] = acc[e];
    __syncthreads();

    if (tid < 64) {
        const int b = rb + tid;
        float l[NE];
        #pragma unroll
        for (int e = 0; e < NE; e++)
            l[e] = part[0][tid][e] + part[1][tid][e] + part[2][tid][e] +
                   part[3][tid][e] + bp[e];
        // sigmoid is monotone -> argmax of logits == argmax of g
        float mx = l[0];
        #pragma unroll
        for (int e = 1; e < NE; e++) mx = fmaxf(mx, l[e]);
        int ties = 0;
        #pragma unroll
        for (int e = 0; e < NE; e++) ties += (l[e] == mx) ? 1 : 0;
        const int flag = (ties > 1) ? TIE_FLAG : 0;
        #pragma unroll
        for (int e = 0; e < NE; e++) {
            if (l[e] == mx) {
                const int pos = atomicAdd(&counts[e], 1);
                list[e * NB + pos] = b | flag;
            }
        }
        if (flag) {  // multi-hot row: GEMM will accumulate atomically
            for (int c = 0; c < NO; c++) out[(size_t)b * NO + c] = 0.f;
        }
    }
}

// ---------------------------------------------------------------- kernel 2
// grid = (128 tiles, 8 experts), 256 threads = 8 waves.
// Block gathers 32 routed rows (2 M-tiles) into LDS; each wave owns two
// 16-wide N-tiles and reuses its B fragments across both M-tiles:
// 4 x v_wmma_f32_16x16x4_f32 per K-step per wave, K=256.
#define XS_PITCH 260  // 260 % 64 == 4 -> distinct banks across 16 rows

__global__ __launch_bounds__(256) void moe_gemm_kernel(
    const float* __restrict__ x, const float* __restrict__ W,
    const int* __restrict__ counts, const int* __restrict__ list,
    float* __restrict__ out) {
    const int e   = blockIdx.y;
    const int cnt = counts[e];
    const int t0  = blockIdx.x * ROWS_PB;
    if (t0 >= cnt) return;

    __shared__ float xs[ROWS_PB * XS_PITCH];  // 32.5KB
    __shared__ int   rows_s[ROWS_PB];

    const int tid = threadIdx.x;
    if (tid < ROWS_PB)
        rows_s[tid] = (t0 + tid < cnt) ? list[e * NB + t0 + tid] : -1;
    __syncthreads();

    // gather 32 rows x 256 cols of x into LDS (zeros for padding rows)
    for (int q = tid; q < ROWS_PB * 64; q += 256) {  // 2048 float4
        const int r  = q >> 6;  // 64 float4 per row
        const int cw = q & 63;
        const int entry = rows_s[r];
        float4 v = make_float4(0.f, 0.f, 0.f, 0.f);
        if (entry >= 0) {
            const int row = entry & 0xFFFF;
            v = *(const float4*)(x + (size_t)row * NI + cw * 4);
        }
        float* d = &xs[r * XS_PITCH + cw * 4];
        d[0] = v.x; d[1] = v.y; d[2] = v.z; d[3] = v.w;
    }
    __syncthreads();

    const int lane  = tid & 31;
    const int wave  = tid >> 5;
    const int m     = lane & 15;            // A-frag row within M-tile
    const int khalf = (lane >> 4) << 1;     // A/B frag K sub-offset {0,2}
    const int nlo   = lane & 15;            // B/C frag column within tile
    const float* We = W + (size_t)e * NI * NO;
    const int col0  = wave * 32 + nlo;      // wave owns N-tiles 2w, 2w+1
    const int col1  = col0 + 16;

    v8f c00 = {}, c01 = {};                 // M-tile 0 x {N-tile0, N-tile1}
    v8f c10 = {}, c11 = {};                 // M-tile 1 x {N-tile0, N-tile1}
    #pragma unroll 2
    for (int k0 = 0; k0 < NI; k0 += 4) {
        const int kk = k0 + khalf;  // even -> 8B-aligned LDS load
        const v2f a0 = *(const v2f*)(&xs[m * XS_PITCH + kk]);
        const v2f a1 = *(const v2f*)(&xs[(m + 16) * XS_PITCH + kk]);
        v2f b0, b1;
        b0.x = We[(size_t)kk * NO + col0];
        b0.y = We[(size_t)(kk + 1) * NO + col0];
        b1.x = We[(size_t)kk * NO + col1];
        b1.y = We[(size_t)(kk + 1) * NO + col1];
        c00 = __builtin_amdgcn_wmma_f32_16x16x4_f32(
            false, a0, false, b0, (short)0, c00, false, false);
        c01 = __builtin_amdgcn_wmma_f32_16x16x4_f32(
            false, a0, false, b1, (short)0, c01, false, false);
        c10 = __builtin_amdgcn_wmma_f32_16x16x4_f32(
            false, a1, false, b0, (short)0, c10, false, false);
        c11 = __builtin_amdgcn_wmma_f32_16x16x4_f32(
            false, a1, false, b1, (short)0, c11, false, false);
    }

    // C/D layout: VGPR v -> M = v + 8*(lane>=16), N = lane%16
    const int mhi = (lane >> 4) << 3;
    #pragma unroll
    for (int v = 0; v < 8; v++) {
        const int M0 = v + mhi;       // row in M-tile 0
        const int M1 = M0 + 16;       // row in M-tile 1
        const int e0 = rows_s[M0];
        const int e1 = rows_s[M1];
        if (e0 >= 0) {
            const int row = e0 & 0xFFFF;
            float* p0 = out + (size_t)row * NO + col0;
            if (e0 & TIE_FLAG) {      // tied row appears in multiple lists
                atomicAdd(p0, c00[v]);
                atomicAdd(p0 + 16, c01[v]);
            } else {
                p0[0]  = c00[v];
                p0[16] = c01[v];
            }
        }
        if (e1 >= 0) {
            const int row = e1 & 0xFFFF;
            float* p1 = out + (size_t)row * NO + col0;
            if (e1 & TIE_FLAG) {
                atomicAdd(p1, c10[v]);
                atomicAdd(p1 + 16, c11[v]);
            } else {
                p1[0]  = c10[v];
                p1[16] = c11[v];
            }
        }
    }
}

// ---------------------------------------------------------------- launch
extern "C" void kernel_launch(void* const* d_in, const int* in_sizes, int n_in,
                              void* d_out, int out_size, void* d_ws,
                              size_t ws_size, hipStream_t stream) {
    const float* x  = (const float*)d_in[0];  // [4096, 256]
    const float* W  = (const float*)d_in[1];  // [8, 256, 256]
    const float* Wp = (const float*)d_in[2];  // [8, 256]
    const float* bp = (const float*)d_in[3];  // [8]
    float* out = (float*)d_out;               // [4096, 256]

    int* counts = (int*)d_ws;                 // 8 ints (64B slot)
    int* list   = (int*)((char*)d_ws + 64);   // 8 * 4096 ints = 128KB

    zero_counts<<<1, 32, 0, stream>>>(counts);
    router_kernel<<<NB / 64, 256, 0, stream>>>(x, Wp, bp, counts, list, out);
    moe_gemm_kernel<<<dim3(NB / ROWS_PB, NE), 256, 0, stream>>>(x, W, counts,
                                                                list, out);
}